// GAT_30846455120748
// MI455X (gfx1250) — compile-verified
//
#include <hip/hip_runtime.h>
#include <math.h>

#define N_NODES 50000
#define N_EDGES 800000
#define E_TOT   (N_EDGES + N_NODES)   // 850000 (self-loops appended)
#define NFEAT   128
#define NHID    8
#define NHEADS  8
#define HID1    (NHEADS * NHID)       // 64
#define NCLASS  40
#define NEG_SLOPE 0.2f

typedef __attribute__((ext_vector_type(2))) float v2f;
typedef __attribute__((ext_vector_type(8))) float v8f;

// ---------------------------------------------------------------------------
// helpers
// ---------------------------------------------------------------------------
__device__ __forceinline__ float lrelu(float x) { return x > 0.f ? x : NEG_SLOPE * x; }

// order-preserving float<->uint mapping so we can use integer atomicMax for
// the segment-max (GLOBAL_ATOMIC_MAX_U32 on CDNA5, done in L2).
__device__ __forceinline__ unsigned f2ord(float f) {
  unsigned u = __float_as_uint(f);
  return (u & 0x80000000u) ? ~u : (u | 0x80000000u);
}
__device__ __forceinline__ float ord2f(unsigned u) {
  return (u & 0x80000000u) ? __uint_as_float(u & 0x7FFFFFFFu) : __uint_as_float(~u);
}
#define ORD_NEG_INF 0x007FFFFFu  // f2ord(-INFINITY)

__device__ __forceinline__ void get_edge(const int* __restrict__ ei, int e, int& s, int& d) {
  if (e < N_EDGES) { s = ei[e]; d = ei[N_EDGES + e]; }
  else             { s = e - N_EDGES; d = s; }        // appended self-loop
}

// ---------------------------------------------------------------------------
// init: zero accumulator region, seed max buffers with ordered(-inf)
// ---------------------------------------------------------------------------
__global__ void init_kernel(float* __restrict__ zbase, unsigned zn,
                            unsigned* __restrict__ mbase, unsigned mn) {
  unsigned i = blockIdx.x * blockDim.x + threadIdx.x;
  if (i < zn) zbase[i] = 0.f;
  if (i < mn) mbase[i] = ORD_NEG_INF;
}

// ---------------------------------------------------------------------------
// GEMM1: H1[50000,64] = X[50000,128] @ W1[128,64]   (V_WMMA_F32_16X16X4_F32)
// One wave per 16-row tile; 4 accumulator tiles cover all 64 output columns.
// f32 16x4 A layout: lanes 0-15 -> K={k0,k0+1}, lanes 16-31 -> K={k0+2,k0+3}.
// ---------------------------------------------------------------------------
__global__ void __launch_bounds__(128) gemm1_wmma(const float* __restrict__ X,
                                                  const float* __restrict__ W,
                                                  float* __restrict__ H) {
  const int wave = threadIdx.x >> 5;
  const int lane = threadIdx.x & 31;
  const int mt   = blockIdx.x * 4 + wave;
  if (mt >= N_NODES / 16) return;                    // 3125 tiles exactly
  const int m  = lane & 15;
  const int hi = lane >> 4;
  const float* xr = X + (long)(mt * 16 + m) * NFEAT;

  v8f c0 = {}, c1 = {}, c2 = {}, c3 = {};
  #pragma unroll 4
  for (int k0 = 0; k0 < NFEAT; k0 += 4) {
    const int ka = k0 + 2 * hi;
    v2f a; a.x = xr[ka]; a.y = xr[ka + 1];
    const float* w0 = W + (long)ka * HID1 + m;       // row ka
    const float* w1 = w0 + HID1;                     // row ka+1
    v2f b;
    b.x = w0[0];  b.y = w1[0];
    c0 = __builtin_amdgcn_wmma_f32_16x16x4_f32(false, a, false, b, (short)0, c0, false, false);
    b.x = w0[16]; b.y = w1[16];
    c1 = __builtin_amdgcn_wmma_f32_16x16x4_f32(false, a, false, b, (short)0, c1, false, false);
    b.x = w0[32]; b.y = w1[32];
    c2 = __builtin_amdgcn_wmma_f32_16x16x4_f32(false, a, false, b, (short)0, c2, false, false);
    b.x = w0[48]; b.y = w1[48];
    c3 = __builtin_amdgcn_wmma_f32_16x16x4_f32(false, a, false, b, (short)0, c3, false, false);
  }
  // C layout: VGPR j -> row (j + 8*hi), col = lane&15
  float* hp = H + (long)(mt * 16 + 8 * hi) * HID1 + m;
  #pragma unroll
  for (int j = 0; j < 8; j++) {
    hp[(long)j * HID1 +  0] = c0[j];
    hp[(long)j * HID1 + 16] = c1[j];
    hp[(long)j * HID1 + 32] = c2[j];
    hp[(long)j * HID1 + 48] = c3[j];
  }
}

// ---------------------------------------------------------------------------
// GEMM2: H2[50000,40] = X2[50000,64] @ W2[64,40]  (pad N to 48, 3 tiles)
// ---------------------------------------------------------------------------
__global__ void __launch_bounds__(128) gemm2_wmma(const float* __restrict__ X,
                                                  const float* __restrict__ W,
                                                  float* __restrict__ H) {
  const int wave = threadIdx.x >> 5;
  const int lane = threadIdx.x & 31;
  const int mt   = blockIdx.x * 4 + wave;
  if (mt >= N_NODES / 16) return;
  const int m  = lane & 15;
  const int hi = lane >> 4;
  const float* xr = X + (long)(mt * 16 + m) * HID1;

  v8f c0 = {}, c1 = {}, c2 = {};
  #pragma unroll 4
  for (int k0 = 0; k0 < HID1; k0 += 4) {
    const int ka = k0 + 2 * hi;
    v2f a; a.x = xr[ka]; a.y = xr[ka + 1];
    #pragma unroll
    for (int t = 0; t < 3; t++) {
      const int col = t * 16 + m;
      v2f b;
      b.x = (col < NCLASS) ? W[(long)ka * NCLASS + col]       : 0.f;
      b.y = (col < NCLASS) ? W[(long)(ka + 1) * NCLASS + col] : 0.f;
      if (t == 0) c0 = __builtin_amdgcn_wmma_f32_16x16x4_f32(false, a, false, b, (short)0, c0, false, false);
      if (t == 1) c1 = __builtin_amdgcn_wmma_f32_16x16x4_f32(false, a, false, b, (short)0, c1, false, false);
      if (t == 2) c2 = __builtin_amdgcn_wmma_f32_16x16x4_f32(false, a, false, b, (short)0, c2, false, false);
    }
  }
  const int rbase = mt * 16 + 8 * hi;
  #pragma unroll
  for (int j = 0; j < 8; j++) {
    float* hr = H + (long)(rbase + j) * NCLASS;
    if (m      < NCLASS) hr[m]      = c0[j];
    if (m + 16 < NCLASS) hr[m + 16] = c1[j];
    if (m + 32 < NCLASS) hr[m + 32] = c2[j];
  }
}

// ---------------------------------------------------------------------------
// per-node attention logits
// ---------------------------------------------------------------------------
__global__ void alpha1_kernel(const float* __restrict__ H1,
                              const float* __restrict__ a_s, const float* __restrict__ a_d,
                              float* __restrict__ as, float* __restrict__ ad) {
  int i = blockIdx.x * blockDim.x + threadIdx.x;      // i = n*8 + h
  if (i >= N_NODES * NHEADS) return;
  const int h = i & 7;
  const float* hp = H1 + (long)(i >> 3) * HID1 + h * NHID;
  float s = 0.f, d = 0.f;
  #pragma unroll
  for (int k = 0; k < NHID; k++) { s += hp[k] * a_s[h * NHID + k]; d += hp[k] * a_d[h * NHID + k]; }
  as[i] = s; ad[i] = d;
}

__global__ void alpha2_kernel(const float* __restrict__ H2,
                              const float* __restrict__ a_s, const float* __restrict__ a_d,
                              float* __restrict__ as, float* __restrict__ ad) {
  int n = blockIdx.x * blockDim.x + threadIdx.x;
  if (n >= N_NODES) return;
  const float* hp = H2 + (long)n * NCLASS;
  float s = 0.f, d = 0.f;
  #pragma unroll
  for (int c = 0; c < NCLASS; c++) { s += hp[c] * a_s[c]; d += hp[c] * a_d[c]; }
  as[n] = s; ad[n] = d;
}

// ---------------------------------------------------------------------------
// layer-1 edge passes (8 heads)
// ---------------------------------------------------------------------------
__global__ void edge_max1(const int* __restrict__ ei, const float* __restrict__ as,
                          const float* __restrict__ ad, unsigned* __restrict__ m) {
  int e = blockIdx.x * blockDim.x + threadIdx.x;
  if (e >= E_TOT) return;
  int s, d; get_edge(ei, e, s, d);
  #pragma unroll
  for (int h = 0; h < NHEADS; h++) {
    float v = lrelu(as[s * NHEADS + h] + ad[d * NHEADS + h]);
    atomicMax(&m[d * NHEADS + h], f2ord(v));
  }
}

__global__ void edge_sum1(const int* __restrict__ ei, const float* __restrict__ as,
                          const float* __restrict__ ad, const unsigned* __restrict__ m,
                          float* __restrict__ den) {
  int e = blockIdx.x * blockDim.x + threadIdx.x;
  if (e >= E_TOT) return;
  int s, d; get_edge(ei, e, s, d);
  #pragma unroll
  for (int h = 0; h < NHEADS; h++) {
    float v = lrelu(as[s * NHEADS + h] + ad[d * NHEADS + h]);
    atomicAdd(&den[d * NHEADS + h], __expf(v - ord2f(m[d * NHEADS + h])));
  }
}

__global__ void edge_agg1(const int* __restrict__ ei, const float* __restrict__ as,
                          const float* __restrict__ ad, const unsigned* __restrict__ m,
                          const float* __restrict__ den, const float* __restrict__ H1,
                          float* __restrict__ out1) {
  int idx = blockIdx.x * blockDim.x + threadIdx.x;    // (edge, head)
  if (idx >= E_TOT * NHEADS) return;
  const int e = idx >> 3, h = idx & 7;
  int s, d; get_edge(ei, e, s, d);
  float v    = lrelu(as[s * NHEADS + h] + ad[d * NHEADS + h]);
  float attn = __expf(v - ord2f(m[d * NHEADS + h])) / den[d * NHEADS + h];
  const float* hs = H1 + (long)s * HID1 + h * NHID;
  float*       od = out1 + (long)d * HID1 + h * NHID;
  #pragma unroll
  for (int k = 0; k < NHID; k++) atomicAdd(&od[k], attn * hs[k]);
}

__global__ void elu_bias_kernel(float* __restrict__ o, const float* __restrict__ b) {
  int i = blockIdx.x * blockDim.x + threadIdx.x;
  if (i >= N_NODES * HID1) return;
  float v = o[i] + b[i & (HID1 - 1)];
  o[i] = v > 0.f ? v : (__expf(v) - 1.f);             // ELU alpha=1, in place
}

// ---------------------------------------------------------------------------
// layer-2 edge passes (1 head, 40 classes)
// ---------------------------------------------------------------------------
__global__ void edge_max2(const int* __restrict__ ei, const float* __restrict__ as,
                          const float* __restrict__ ad, unsigned* __restrict__ m) {
  int e = blockIdx.x * blockDim.x + threadIdx.x;
  if (e >= E_TOT) return;
  int s, d; get_edge(ei, e, s, d);
  atomicMax(&m[d], f2ord(lrelu(as[s] + ad[d])));
}

__global__ void edge_sum2(const int* __restrict__ ei, const float* __restrict__ as,
                          const float* __restrict__ ad, const unsigned* __restrict__ m,
                          float* __restrict__ den) {
  int e = blockIdx.x * blockDim.x + threadIdx.x;
  if (e >= E_TOT) return;
  int s, d; get_edge(ei, e, s, d);
  float v = lrelu(as[s] + ad[d]);
  atomicAdd(&den[d], __expf(v - ord2f(m[d])));
}

__global__ void edge_agg2(const int* __restrict__ ei, const float* __restrict__ as,
                          const float* __restrict__ ad, const unsigned* __restrict__ m,
                          const float* __restrict__ den, const float* __restrict__ H2,
                          float* __restrict__ out2) {
  int e = blockIdx.x * blockDim.x + threadIdx.x;
  if (e >= E_TOT) return;
  int s, d; get_edge(ei, e, s, d);
  float v    = lrelu(as[s] + ad[d]);
  float attn = __expf(v - ord2f(m[d])) / den[d];
  const float* hs = H2 + (long)s * NCLASS;
  float*       od = out2 + (long)d * NCLASS;
  #pragma unroll
  for (int c = 0; c < NCLASS; c++) atomicAdd(&od[c], attn * hs[c]);
}

// ---------------------------------------------------------------------------
// final: out = log_softmax(out2 + b2)
// ---------------------------------------------------------------------------
__global__ void lsm_kernel(const float* __restrict__ o2, const float* __restrict__ b,
                           float* __restrict__ out) {
  int n = blockIdx.x * blockDim.x + threadIdx.x;
  if (n >= N_NODES) return;
  float buf[NCLASS];
  float mx = -INFINITY;
  #pragma unroll
  for (int c = 0; c < NCLASS; c++) { buf[c] = o2[(long)n * NCLASS + c] + b[c]; mx = fmaxf(mx, buf[c]); }
  float ssum = 0.f;
  #pragma unroll
  for (int c = 0; c < NCLASS; c++) ssum += __expf(buf[c] - mx);
  const float lse = mx + __logf(ssum);
  #pragma unroll
  for (int c = 0; c < NCLASS; c++) out[(long)n * NCLASS + c] = buf[c] - lse;
}

// ---------------------------------------------------------------------------
// launcher
// ---------------------------------------------------------------------------
extern "C" void kernel_launch(void* const* d_in, const int* in_sizes, int n_in,
                              void* d_out, int out_size, void* d_ws, size_t ws_size,
                              hipStream_t stream) {
  const float* x      = (const float*)d_in[0];
  const float* W1     = (const float*)d_in[1];
  const float* a_src1 = (const float*)d_in[2];
  const float* a_dst1 = (const float*)d_in[3];
  const float* b1     = (const float*)d_in[4];
  const float* W2     = (const float*)d_in[5];
  const float* a_src2 = (const float*)d_in[6];
  const float* a_dst2 = (const float*)d_in[7];
  const float* b2     = (const float*)d_in[8];
  const int*   ei     = (const int*)d_in[9];
  float*       out    = (float*)d_out;

  // workspace carve-up (floats)
  float* ws   = (float*)d_ws;
  float* H1   = ws;                                  // 3,200,000
  float* as1  = H1  + (long)N_NODES * HID1;          //   400,000
  float* ad1  = as1 + (long)N_NODES * NHEADS;        //   400,000
  float* as2  = ad1 + (long)N_NODES * NHEADS;        //    50,000
  float* ad2  = as2 + N_NODES;                       //    50,000
  float* zb   = ad2 + N_NODES;                       // zero-region start
  float* den1 = zb;                                  //   400,000
  float* out1 = den1 + (long)N_NODES * NHEADS;       // 3,200,000 (X2 after ELU)
  float* den2 = out1 + (long)N_NODES * HID1;         //    50,000
  float* out2 = den2 + N_NODES;                      // 2,000,000
  const unsigned ZERON = N_NODES * NHEADS + N_NODES * HID1 + N_NODES + N_NODES * NCLASS;
  unsigned* m1 = (unsigned*)(out2 + (long)N_NODES * NCLASS); // 400,000
  unsigned* m2 = m1 + (long)N_NODES * NHEADS;                //  50,000
  const unsigned MINFN = N_NODES * NHEADS + N_NODES;
  float* H2   = (float*)(m2 + N_NODES);              // 2,000,000

  const int B = 256;
  auto cdiv = [](long a, long b) { return (int)((a + b - 1) / b); };

  init_kernel<<<cdiv(ZERON, B), B, 0, stream>>>(zb, ZERON, m1, MINFN);

  // ---- layer 1 ----
  gemm1_wmma<<<cdiv(N_NODES / 16, 4), 128, 0, stream>>>(x, W1, H1);
  alpha1_kernel<<<cdiv((long)N_NODES * NHEADS, B), B, 0, stream>>>(H1, a_src1, a_dst1, as1, ad1);
  edge_max1<<<cdiv(E_TOT, B), B, 0, stream>>>(ei, as1, ad1, m1);
  edge_sum1<<<cdiv(E_TOT, B), B, 0, stream>>>(ei, as1, ad1, m1, den1);
  edge_agg1<<<cdiv((long)E_TOT * NHEADS, B), B, 0, stream>>>(ei, as1, ad1, m1, den1, H1, out1);
  elu_bias_kernel<<<cdiv((long)N_NODES * HID1, B), B, 0, stream>>>(out1, b1);

  // ---- layer 2 ----
  gemm2_wmma<<<cdiv(N_NODES / 16, 4), 128, 0, stream>>>(out1, W2, H2);
  alpha2_kernel<<<cdiv(N_NODES, B), B, 0, stream>>>(H2, a_src2, a_dst2, as2, ad2);
  edge_max2<<<cdiv(E_TOT, B), B, 0, stream>>>(ei, as2, ad2, m2);
  edge_sum2<<<cdiv(E_TOT, B), B, 0, stream>>>(ei, as2, ad2, m2, den2);
  edge_agg2<<<cdiv(E_TOT, B), B, 0, stream>>>(ei, as2, ad2, m2, den2, H2, out2);

  lsm_kernel<<<cdiv(N_NODES, B), B, 0, stream>>>(out2, b2, out);
}